// DistCrossEntropyLoss_49486613184859
// MI455X (gfx1250) — compile-verified
//
#include <hip/hip_runtime.h>

// CDNA5 / gfx1250, wave32. Fused cross-entropy + softmax-Hessian contraction.
// out[b] = logsumexp(m[b]) - m[b,i[b]] + 0.5*( sum_j k[b,j,j] p_j  -  p^T K p )
// One 256-thread block (8 wave32) per batch; k[b] (256 KB) streamed exactly once.
//
// quad = p^T K p via V_WMMA_F32_16X16X4_F32:
//   C(16x16) += A(16x4 tile of K) x B(4x16),  B[kk,n] = p[j+kk] replicated over n
//   => C[i,n] = (K p)_i for all n; wave-sum of p_i*C[i,n] over all lanes = 16*quad_block.

typedef __attribute__((ext_vector_type(2))) float v2f;
typedef __attribute__((ext_vector_type(8))) float v8f;

#define CE_D 256   // feature dim (fixed by reference)

__device__ __forceinline__ float block_reduce_sum(float v, float* s, int t) {
  __syncthreads();                 // protect previous use of s
  s[t] = v;
  __syncthreads();
  #pragma unroll
  for (int off = 128; off > 0; off >>= 1) {
    if (t < off) s[t] += s[t + off];
    __syncthreads();
  }
  float r = s[0];
  __syncthreads();
  return r;
}

__device__ __forceinline__ float block_reduce_max(float v, float* s, int t) {
  __syncthreads();
  s[t] = v;
  __syncthreads();
  #pragma unroll
  for (int off = 128; off > 0; off >>= 1) {
    if (t < off) s[t] = fmaxf(s[t], s[t + off]);
    __syncthreads();
  }
  float r = s[0];
  __syncthreads();
  return r;
}

__global__ __launch_bounds__(256)
void ce_softmax_hessian_kernel(const float* __restrict__ m,
                               const float* __restrict__ k,
                               const int*   __restrict__ idx,
                               float*       __restrict__ out) {
  constexpr int D = CE_D;
  __shared__ float s_p[D];       // softmax probabilities
  __shared__ float s_red[256];   // reduction scratch

  const int b = blockIdx.x;
  const int t = threadIdx.x;                  // 0..255, one lane per feature
  const float* mb = m + (size_t)b * D;
  const float* kb = k + (size_t)b * D * D;

  // ---------------- softmax over m[b,:] ----------------
  const float mv  = mb[t];
  const float mx  = block_reduce_max(mv, s_red, t);
  const float e   = __expf(mv - mx);
  const float sum = block_reduce_sum(e, s_red, t);
  const float p   = e / sum;
  s_p[t] = p;

  // ---------------- trace term: sum_j k[b,j,j] * p_j ----------------
  const float trace = block_reduce_sum(kb[(size_t)t * (D + 1)] * p, s_red, t);
  // (the barriers inside block_reduce_sum also make s_p[] visible to all waves)

  // ---------------- quad = p^T K p via WMMA f32 16x16x4 ----------------
  const int lane   = t & 31;
  const int wave   = t >> 5;        // 0..7
  const int lane15 = lane & 15;
  const int hi     = lane >> 4;     // ISA 16x4 A layout: lo lanes K=j,j+1; hi lanes K=j+2,j+3

  // Per-lane B source: p[j+2*hi], p[j+2*hi+1]  (replicated over all N columns,
  // so the N->lane mapping of the B operand is irrelevant; only the K-split
  // matters, which mirrors the documented A-operand layout).
  const float* pb = s_p + 2 * hi;

  float quad_partial = 0.0f;
  #pragma unroll
  for (int ib = 0; ib < 2; ++ib) {
    const int I   = wave + ib * 8;               // 16-row block of K, 0..15
    const int row = I * 16 + lane15;
    const float* arow = kb + (size_t)row * D + 2 * hi;   // contiguous float2 per lane

    v8f acc[4] = {};   // 4 rotating accumulators to break WMMA->WMMA RAW chain
    #pragma unroll 16
    for (int j = 0; j < D; j += 4) {
      v2f a  = *(const v2f*)(arow + j);          // one global_load_b64 per lane
      v2f bv = *(const v2f*)(pb + j);            // one ds_load_b64 per lane (LDS)
      const int sel = (j >> 2) & 3;
      acc[sel] = __builtin_amdgcn_wmma_f32_16x16x4_f32(
          /*neg_a=*/false, a, /*neg_b=*/false, bv,
          /*c_mod=*/(short)0, acc[sel], /*reuse_a=*/false, /*reuse_b=*/false);
    }
    v8f accs = (acc[0] + acc[1]) + (acc[2] + acc[3]);
    // C/D layout: lanes 0-15: VGPR v -> row M=v; lanes 16-31: VGPR v -> row M=v+8.
    #pragma unroll
    for (int v = 0; v < 8; ++v)
      quad_partial += s_p[I * 16 + hi * 8 + v] * accs[v];
  }
  quad_partial *= 0.0625f;   // every row counted once per each of 16 B-columns

  const float quad = block_reduce_sum(quad_partial, s_red, t);

  if (t == 0) {
    const float lse    = mx + __logf(sum);
    const float picked = mb[idx[b]];
    out[b] = (lse - picked) + 0.5f * (trace - quad);
  }
}

extern "C" void kernel_launch(void* const* d_in, const int* in_sizes, int n_in,
                              void* d_out, int out_size, void* d_ws, size_t ws_size,
                              hipStream_t stream) {
  const float* m   = (const float*)d_in[0];   // [B, D] fp32
  const float* k   = (const float*)d_in[1];   // [B, D, D] fp32
  const int*   idx = (const int*)d_in[2];     // [B] int
  float* out = (float*)d_out;                 // [B] fp32

  const int B = in_sizes[2];                  // 2048 batches, one block each
  ce_softmax_hessian_kernel<<<dim3(B), dim3(256), 0, stream>>>(m, k, idx, out);

  (void)n_in; (void)out_size; (void)d_ws; (void)ws_size;
}